// HierarchicalNTXentLoss_52080773431812
// MI455X (gfx1250) — compile-verified
//
#include <hip/hip_runtime.h>
#include <math.h>

typedef float v2f __attribute__((ext_vector_type(2)));
typedef float v8f __attribute__((ext_vector_type(8)));
typedef int   v4i_vs __attribute__((vector_size(16)));   // matches builtin param

#define BB 4096
#define NN 8192
#define DD 128
#define INV_T 20.0f

#define NSPLIT 4
#define CT_TOTAL (NN / 16)              /* 512 column tiles  */
#define CT_PER   (CT_TOTAL / NSPLIT)    /* 128 per block     */
#define BSTRIDE  132                    /* padded col stride (dwords): bank-conflict-free */

#if __has_builtin(__builtin_amdgcn_global_load_async_to_lds_b128) && \
    __has_builtin(__builtin_amdgcn_s_wait_asynccnt)
#define ASYNC_LDS 1
#endif

__device__ __forceinline__ void stage16(const float* g, float* l)
{
#ifdef ASYNC_LDS
    __builtin_amdgcn_global_load_async_to_lds_b128(
        (__attribute__((address_space(1))) v4i_vs*)g,
        (__attribute__((address_space(3))) v4i_vs*)l, 0, 0);
#else
    *(float4*)l = *(const float4*)g;
#endif
}

__device__ __forceinline__ void stage_wait()
{
#ifdef ASYNC_LDS
    __builtin_amdgcn_s_wait_asynccnt(0);
#endif
}

// ---------------------------------------------------------------------------
// Kernel 1: row-normalize z1|z2 into zall, pack labels into 4 bits
// ---------------------------------------------------------------------------
__global__ __launch_bounds__(128) void k_normalize(
    const float* __restrict__ z1, const float* __restrict__ z2,
    const int* __restrict__ m1, const int* __restrict__ m2,
    const int* __restrict__ c1, const int* __restrict__ c2,
    const int* __restrict__ p1, const int* __restrict__ p2,
    float* __restrict__ zall, int* __restrict__ labels)
{
    __shared__ float red[128];
    const int row = blockIdx.x;
    const int t   = threadIdx.x;
    const float* src = (row < BB) ? (z1 + (size_t)row * DD)
                                  : (z2 + (size_t)(row - BB) * DD);
    const float v = src[t];
    red[t] = v * v;
    __syncthreads();
    for (int s = 64; s > 0; s >>= 1) {
        if (t < s) red[t] += red[t + s];
        __syncthreads();
    }
    const float inv = 1.0f / sqrtf(red[0]);
    zall[(size_t)row * DD + t] = v * inv;
    if (t == 0) {
        const int m = (row < BB) ? m1[row] : m2[row - BB];
        const int c = (row < BB) ? c1[row] : c2[row - BB];
        const int p = (row < BB) ? p1[row] : p2[row - BB];
        labels[row] = (m & 3) | ((c & 1) << 2) | ((p & 1) << 3);
    }
}

// ---------------------------------------------------------------------------
// Kernel 2: per-row pos[i] = 20 * <z_i, z_partner>, max_h[i] over negatives
// ---------------------------------------------------------------------------
__global__ __launch_bounds__(128) void k_rowstats(
    const float* __restrict__ zall, const int* __restrict__ labels,
    float* __restrict__ maxh, float* __restrict__ pos)
{
    __shared__ float redS[128];
    __shared__ float redM[128];
    const int i = blockIdx.x;
    const int t = threadIdx.x;
    const int part = i ^ BB;
    const float d = zall[(size_t)i * DD + t] * zall[(size_t)part * DD + t];
    const int li = labels[i];
    float hm = 0.0f;
    for (int j = t; j < NN; j += 128) {
        if (j == i || j == part) continue;
        const int x = li ^ labels[j];
        float H = 0.0f;
        if ((x & 3) == 0) {
            H = 1.0f;
            if ((x & 4) == 0) H += 0.5f;
            if ((x & 8) == 0) H += 0.25f;
        }
        hm = fmaxf(hm, H);
    }
    redS[t] = d; redM[t] = hm;
    __syncthreads();
    for (int s = 64; s > 0; s >>= 1) {
        if (t < s) { redS[t] += redS[t + s]; redM[t] = fmaxf(redM[t], redM[t + s]); }
        __syncthreads();
    }
    if (t == 0) { pos[i] = redS[0] * INV_T; maxh[i] = redM[0]; }
}

// ---------------------------------------------------------------------------
// Kernel 3: fused fp32-WMMA GEMM + weighted sum-exp.
// Grid (64, NSPLIT): block = 8 waves, 128 rows; waves share B tiles staged
// into double-buffered LDS via async global->LDS, padded to 132-dword column
// stride (b64 reads cover all 64 banks exactly once).
// ---------------------------------------------------------------------------
__global__ __launch_bounds__(256) void k_main(
    const float* __restrict__ zall, const int* __restrict__ labels,
    const float* __restrict__ maxh, float* __restrict__ parts)
{
    __shared__ float bbuf[2][16 * BSTRIDE];

    const int split = blockIdx.y;
    const int ct0   = split * CT_PER;
    const int tid   = threadIdx.x;
    const int wave  = tid >> 5;
    const int lane  = tid & 31;
    const int half  = lane >> 4;        // K sub-pair / M half selector
    const int l15   = lane & 15;        // M for A, N for B/C
    const int moff  = half << 3;
    const int rowBase = blockIdx.x * 128 + wave * 16;

    // A fragments: a[s] = { z[row][4s + 2*half], z[row][4s + 2*half + 1] }
    const float* aptr = zall + (size_t)(rowBase + l15) * DD + (half << 1);
    v2f a[32];
#pragma unroll
    for (int s = 0; s < 32; ++s) {
        const float2 t = *(const float2*)(aptr + 4 * s);
        a[s].x = t.x; a[s].y = t.y;
    }

    int   lrow[8];
    float mh8[8];
    float srun[8];
#pragma unroll
    for (int r = 0; r < 8; ++r) {
        lrow[r] = labels[rowBase + moff + r];
        mh8[r]  = maxh[rowBase + moff + r];
        srun[r] = 0.0f;
    }

    // staging: thread -> (col 0..15, chunk 0..15 of 8 dwords)
    const int scol   = tid >> 4;
    const int schunk = tid & 15;
    float* l0 = &bbuf[0][scol * BSTRIDE + schunk * 8];
    float* l1 = &bbuf[1][scol * BSTRIDE + schunk * 8];

    // prologue: stage tile ct0 into buffer 0
    {
        const float* g = zall + (size_t)(ct0 * 16 + scol) * DD + schunk * 8;
        stage16(g, l0);
        stage16(g + 4, l0 + 4);
    }

    for (int it = 0; it < CT_PER; ++it) {
        const int ct = ct0 + it;
        stage_wait();               // my async writes to buf[it&1] complete
        __syncthreads();            // everyone's writes done; prev reads done

        if (it + 1 < CT_PER) {      // uniform: prefetch next tile
            const float* g = zall + (size_t)((ct + 1) * 16 + scol) * DD + schunk * 8;
            float* l = (it & 1) ? l0 : l1;
            stage16(g, l);
            stage16(g + 4, l + 4);
        }

        // compute 16x16 sim tile from LDS buffer (32 x fp32 WMMA, K=4)
        const float* bfrag = &bbuf[it & 1][l15 * BSTRIDE + (half << 1)];
        v8f c = {0.f, 0.f, 0.f, 0.f, 0.f, 0.f, 0.f, 0.f};
#pragma unroll
        for (int s = 0; s < 32; ++s) {
            const float2 t = *(const float2*)(bfrag + 4 * s);   // ds_load_b64
            v2f b; b.x = t.x; b.y = t.y;
            c = __builtin_amdgcn_wmma_f32_16x16x4_f32(
                    false, a[s], false, b, (short)0, c, false, false);
        }

        // epilogue: |sim*w| <= 37 -> exp cannot overflow fp32, plain sum-exp
        const int col = (ct << 4) + l15;
        const int lj  = labels[col];
#pragma unroll
        for (int r = 0; r < 8; ++r) {
            const int i = rowBase + moff + r;
            const int x = lrow[r] ^ lj;
            const float H = ((x & 3) == 0)
                ? (1.0f + (((x & 4) == 0) ? 0.5f : 0.0f)
                        + (((x & 8) == 0) ? 0.25f : 0.0f))
                : 0.0f;
            const float w   = mh8[r] - H + 0.1f;
            const float val = c[r] * INV_T * w;
            float e = __expf(val);
            if (col == i || col == (i ^ BB)) e = 0.0f;   // self / positive
            srun[r] += e;
        }
    }

    // sum across the 16 lanes of each half (xor masks 1..8 stay in-half)
#pragma unroll
    for (int r = 0; r < 8; ++r) {
#pragma unroll
        for (int m = 1; m <= 8; m <<= 1)
            srun[r] += __shfl_xor(srun[r], m, 32);
    }
    if (l15 == 0) {
#pragma unroll
        for (int r = 0; r < 8; ++r)
            parts[(size_t)split * NN + rowBase + moff + r] = srun[r];
    }
}

// ---------------------------------------------------------------------------
// Kernel 4: combine split partials per row, logaddexp, deterministic mean
// ---------------------------------------------------------------------------
__global__ __launch_bounds__(1024) void k_final(
    const float* __restrict__ parts, const float* __restrict__ pos,
    float* __restrict__ out)
{
    __shared__ float red[1024];
    const int t = threadIdx.x;
    float acc = 0.0f;
    for (int row = t; row < NN; row += 1024) {
        float S = 0.0f;
#pragma unroll
        for (int s = 0; s < NSPLIT; ++s) S += parts[(size_t)s * NN + row];
        const float L  = logf(S);
        const float p  = pos[row];
        const float mx = fmaxf(p, L);
        acc += mx + logf(__expf(p - mx) + __expf(L - mx)) - p;
    }
    red[t] = acc;
    __syncthreads();
    for (int s = 512; s > 0; s >>= 1) {
        if (t < s) red[t] += red[t + s];
        __syncthreads();
    }
    if (t == 0) out[0] = red[0] / (float)NN;
}

// ---------------------------------------------------------------------------
extern "C" void kernel_launch(void* const* d_in, const int* in_sizes, int n_in,
                              void* d_out, int out_size, void* d_ws, size_t ws_size,
                              hipStream_t stream) {
    const float* z1 = (const float*)d_in[0];
    const float* z2 = (const float*)d_in[1];
    const int*   m1 = (const int*)d_in[2];
    const int*   m2 = (const int*)d_in[3];
    const int*   c1 = (const int*)d_in[4];
    const int*   c2 = (const int*)d_in[5];
    const int*   p1 = (const int*)d_in[6];
    const int*   p2 = (const int*)d_in[7];
    float* out = (float*)d_out;

    char* base = (char*)d_ws;
    float* zall   = (float*)base;                              // 8192*128 f32
    int*   labels = (int*)(base + (size_t)NN * DD * 4);        // 8192 i32
    float* maxh   = (float*)((char*)labels + (size_t)NN * 4);  // 8192 f32
    float* pos    = (float*)((char*)maxh   + (size_t)NN * 4);  // 8192 f32
    float* parts  = (float*)((char*)pos    + (size_t)NN * 4);  // NSPLIT*8192 f32

    k_normalize<<<NN, 128, 0, stream>>>(z1, z2, m1, m2, c1, c2, p1, p2,
                                        zall, labels);
    k_rowstats<<<NN, 128, 0, stream>>>(zall, labels, maxh, pos);
    dim3 grid(NN / 128, NSPLIT);
    k_main<<<grid, 256, 0, stream>>>(zall, labels, maxh, parts);
    k_final<<<1, 1024, 0, stream>>>(parts, pos, out);
}